// SparseConv3d_14568529068663
// MI455X (gfx1250) — compile-verified
//
#include <hip/hip_runtime.h>

typedef __attribute__((ext_vector_type(2))) float v2f;
typedef __attribute__((ext_vector_type(8))) float v8f;

#define CIN  64
#define COUT 64
#define KOFF 27

// Zero 64-float "feat_pad" dummy row in workspace (graph-capture-safe,
// deterministic: runs every launch).
__global__ void zero_pad_row(float* ws) { ws[threadIdx.x] = 0.0f; }

// One wave32 computes a 16-row x 64-col fp32 output tile using
// V_WMMA_F32_16X16X4_F32 (exact fp32, matching the fp32 reference).
//
// A (16x4 f32): lanes 0-15 = rows M=0..15 holding K=0 (v0), K=1 (v1);
//               lanes 16-31 = same rows holding K=2, K=3.
// B (4x16 f32): lanes 0-15 = cols N=0..15 holding K=0 (v0), K=1 (v1);
//               lanes 16-31 = same cols holding K=2, K=3.
// C/D (16x16 f32): VGPR v, lanes 0-15 -> M=v, N=lane; lanes 16-31 -> M=v+8.
__global__ __launch_bounds__(256) void sparse_conv3d_wmma_f32(
    const float* __restrict__ feat,      // [n][64]
    const float* __restrict__ weight,    // [27][64][64]
    const float* __restrict__ bias,      // [64]
    const int*   __restrict__ rulebook,  // [27][n][2]
    const float* __restrict__ zrow,      // [64] zeros (feat_pad dummy row)
    float*       __restrict__ out,       // [n][64]
    int n, int numTiles)
{
    const int lane   = threadIdx.x & 31;
    const int tile   = (int)((blockIdx.x * blockDim.x + threadIdx.x) >> 5);
    if (tile >= numTiles) return;

    const int  i0     = tile * 16;
    const int  r      = lane & 15;        // tile row (A) / tile col (B,C) for this lane
    const bool hiHalf = lane >= 16;       // lanes 16-31: K+2 slices of A/B, M+8 rows of C
    const int  row    = i0 + r;

    v8f c[4];
    #pragma unroll
    for (int nt = 0; nt < 4; ++nt) c[nt] = v8f{};

    for (int k = 0; k < KOFF; ++k) {
        // Per-lane gather index (read-once stream -> non-temporal).
        int idx = n;
        if (row < n)
            idx = __builtin_nontemporal_load(
                &rulebook[((size_t)k * (size_t)n + (size_t)row) * 2]);
        const bool valid = idx < n;

        // Whole 16-row tile has no neighbor at this offset -> skip 64 WMMAs.
        if (__builtin_amdgcn_ballot_w32(valid) == 0u) continue;

        const float* __restrict__ wk = weight + (size_t)k * (CIN * COUT);
        // Branchless: dummy rows gather from the zeroed pad row (one
        // pointer-select per offset instead of per-K-step exec masking).
        const float* arow = valid ? (feat + (size_t)idx * CIN) : zrow;

        #pragma unroll 4
        for (int k4 = 0; k4 < 16; ++k4) {
            const int kk = k4 * 4 + (hiHalf ? 2 : 0);

            // A fragment: two contiguous fp32 of the gathered row (b64 load).
            const v2f a = *(const v2f*)(arow + kk);

            // Four 16-wide output column tiles share this A fragment.
            #pragma unroll
            for (int nt = 0; nt < 4; ++nt) {
                const float* wp = wk + (size_t)kk * COUT + nt * 16 + r;
                v2f b;
                b.x = wp[0];     // W[k][kk  ][col]
                b.y = wp[COUT];  // W[k][kk+1][col]
                c[nt] = __builtin_amdgcn_wmma_f32_16x16x4_f32(
                    /*neg_a=*/false, a, /*neg_b=*/false, b,
                    /*c_mod=*/(short)0, c[nt],
                    /*reuse_a=*/false, /*reuse_b=*/false);
            }
        }
    }

    // Epilogue: bias add + non-temporal store (output is write-once; keep it
    // out of L2 so features/weights stay resident for the gather/B streams).
    #pragma unroll
    for (int nt = 0; nt < 4; ++nt) {
        const int   col = nt * 16 + r;
        const float bs  = bias[col];
        #pragma unroll
        for (int v = 0; v < 8; ++v) {
            const int m    = v + (hiHalf ? 8 : 0);
            const int orow = i0 + m;
            if (orow < n)
                __builtin_nontemporal_store(c[nt][v] + bs,
                                            &out[(size_t)orow * COUT + col]);
        }
    }
}

extern "C" void kernel_launch(void* const* d_in, const int* in_sizes, int n_in,
                              void* d_out, int out_size, void* d_ws, size_t ws_size,
                              hipStream_t stream) {
    const float* feat     = (const float*)d_in[0];
    const float* weight   = (const float*)d_in[1];
    const float* bias     = (const float*)d_in[2];
    const int*   rulebook = (const int*)d_in[3];
    float*       out      = (float*)d_out;
    float*       zrow     = (float*)d_ws;

    const int n        = in_sizes[0] / CIN;      // 200000
    const int numTiles = (n + 15) / 16;          // 12500 (exact, no tail)

    // Stage 1: zero the feat_pad dummy row in workspace.
    zero_pad_row<<<1, CIN, 0, stream>>>(zrow);

    // Stage 2: main WMMA kernel (8 waves / 256 threads per block).
    const int threads = 256;
    const int blocks  = (numTiles * 32 + threads - 1) / threads;
    sparse_conv3d_wmma_f32<<<blocks, threads, 0, stream>>>(
        feat, weight, bias, rulebook, zrow, out, n, numTiles);
}